// Relation_Classifier_27092653703728
// MI455X (gfx1250) — compile-verified
//
#include <hip/hip_runtime.h>
#include <math.h>

typedef __attribute__((ext_vector_type(2))) float v2f;
typedef __attribute__((ext_vector_type(8))) float v8f;

#define TNF 4  // n-tiles per wave; with 2 m-tiles a wave computes a 32x64 block

// ---------------- block-wide reductions (wave32-aware) ----------------

__device__ __forceinline__ float block_allreduce_sum(float v) {
  __shared__ float sh_sum[9];
  const int lane = threadIdx.x & 31;
  const int w    = threadIdx.x >> 5;
#pragma unroll
  for (int o = 16; o > 0; o >>= 1) v += __shfl_down(v, o, 32);
  __syncthreads();
  if (lane == 0) sh_sum[w] = v;
  __syncthreads();
  if (threadIdx.x == 0) {
    float r = 0.f;
    const int nw = (blockDim.x + 31) >> 5;
    for (int i = 0; i < nw; ++i) r += sh_sum[i];
    sh_sum[8] = r;
  }
  __syncthreads();
  return sh_sum[8];
}

__device__ __forceinline__ float block_allreduce_max(float v) {
  __shared__ float sh_max[9];
  const int lane = threadIdx.x & 31;
  const int w    = threadIdx.x >> 5;
#pragma unroll
  for (int o = 16; o > 0; o >>= 1) v = fmaxf(v, __shfl_down(v, o, 32));
  __syncthreads();
  if (lane == 0) sh_max[w] = v;
  __syncthreads();
  if (threadIdx.x == 0) {
    float r = -3.0e38f;
    const int nw = (blockDim.x + 31) >> 5;
    for (int i = 0; i < nw; ++i) r = fmaxf(r, sh_max[i]);
    sh_max[8] = r;
  }
  __syncthreads();
  return sh_max[8];
}

// ---------------- WMMA f32 16x16x4 wave-level GEMM cores ----------------
// Each wave computes a 32x64 tile of C = A * W^T (A [M,K] rm, W [N,K] rm).
// f32 A-frag: lane L holds A[m0+(L&15)][k + 2*(L>>4) + {0,1}]  (ISA 7.12.2);
// B-frag mirrors it with N, so both load as contiguous float2.
// Out-of-range columns read a zeroed scratch row instead of being masked,
// keeping the inner loop pure load+wmma (no per-iteration cndmask).

__device__ __forceinline__ void wave_gemm_rowrow2(const float* __restrict__ A, long lda,
                                                  const float* __restrict__ W, long ldw,
                                                  const float* __restrict__ zrow,
                                                  int K, int m0, int n0, int Nvalid,
                                                  v8f acc0[TNF], v8f acc1[TNF]) {
  const int lane = threadIdx.x & 31;
  const int kh   = (lane >> 4) << 1;  // 0 or 2
  const float* ap0 = A + (long)(m0 + (lane & 15)) * lda + kh;
  const float* ap1 = ap0 + 16 * lda;
  const float* bp[TNF];
#pragma unroll
  for (int t = 0; t < TNF; ++t) {
    int n = n0 + t * 16 + (lane & 15);
    bp[t] = (n < Nvalid) ? (W + (long)n * ldw + kh) : (zrow + kh);
  }
  for (int k = 0; k < K; k += 4) {
    v2f a0 = *(const v2f*)(ap0 + k);
    v2f a1 = *(const v2f*)(ap1 + k);
#pragma unroll
    for (int t = 0; t < TNF; ++t) {
      v2f b = *(const v2f*)(bp[t] + k);
      acc0[t] = __builtin_amdgcn_wmma_f32_16x16x4_f32(false, a0, false, b,
                                                      (short)0, acc0[t], false, false);
      acc1[t] = __builtin_amdgcn_wmma_f32_16x16x4_f32(false, a1, false, b,
                                                      (short)0, acc1[t], false, false);
    }
  }
}

// C = A * B with B row-major [K,N]; K multiple of 4 (1000 qualifies), no guards.
__device__ __forceinline__ void wave_gemm_rowB2(const float* __restrict__ A, long lda,
                                                const float* __restrict__ B, long ldb,
                                                int K, int m0, int n0,
                                                v8f acc0[TNF], v8f acc1[TNF]) {
  const int lane = threadIdx.x & 31;
  const int kh   = (lane >> 4) << 1;
  const float* ap0 = A + (long)(m0 + (lane & 15)) * lda + kh;
  const float* ap1 = ap0 + 16 * lda;
  int ncol[TNF];
#pragma unroll
  for (int t = 0; t < TNF; ++t) ncol[t] = n0 + t * 16 + (lane & 15);
  for (int k = 0; k < K; k += 4) {
    v2f a0 = *(const v2f*)(ap0 + k);
    v2f a1 = *(const v2f*)(ap1 + k);
    const int k0 = k + kh;
#pragma unroll
    for (int t = 0; t < TNF; ++t) {
      v2f b;
      b[0] = B[(long)k0 * ldb + ncol[t]];
      b[1] = B[(long)(k0 + 1) * ldb + ncol[t]];
      acc0[t] = __builtin_amdgcn_wmma_f32_16x16x4_f32(false, a0, false, b,
                                                      (short)0, acc0[t], false, false);
      acc1[t] = __builtin_amdgcn_wmma_f32_16x16x4_f32(false, a1, false, b,
                                                      (short)0, acc1[t], false, false);
    }
  }
}

// Wave tiling inside a 256-thread block (8 wave32):
//   wave w: m0 = (w&3)*32  (two 16-row tiles), n0 = blockIdx.x*128 + (w>>2)*64.
__device__ __forceinline__ void wave_coords(int& m0, int& n0) {
  const int w = threadIdx.x >> 5;
  m0 = (w & 3) * 32;
  n0 = blockIdx.x * 128 + (w >> 2) * 64;
}

// ---------------- kernels ----------------

// Zero the dummy row and seed the per-row min-dist^2 with +inf bits.
__global__ void init_ws_kernel(float* zrow, int ZN, unsigned* mind2, int B) {
  int i = blockIdx.x * blockDim.x + threadIdx.x;
  if (i < ZN) zrow[i] = 0.f;
  if (i < B) mind2[i] = 0x7f800000u;
}

// mode 0: out = sum(row^2); mode 1: out = 1/sqrt(sum(row^2))
__global__ void rownorm_kernel(const float* __restrict__ src, int D, int mode,
                               float* __restrict__ out) {
  const long off = (long)blockIdx.x * D;
  float s = 0.f;
  for (int i = threadIdx.x; i < D; i += blockDim.x) {
    float v = src[off + i];
    s += v * v;
  }
  s = block_allreduce_sum(s);
  if (threadIdx.x == 0) out[blockIdx.x] = (mode == 0) ? s : rsqrtf(s);
}

// GEMM 1: x @ centers^T fused with min_c ||x-c||^2 via atomicMin on uint bits.
__global__ void gemm_dist_min_kernel(const float* __restrict__ x,
                                     const float* __restrict__ centers,
                                     const float* __restrict__ zrow,
                                     int K, int N,
                                     const float* __restrict__ xs2,
                                     const float* __restrict__ cn2,
                                     unsigned* __restrict__ mind2) {
  int m0, n0;
  wave_coords(m0, n0);
  v8f acc0[TNF] = {}, acc1[TNF] = {};
  wave_gemm_rowrow2(x, K, centers, K, zrow, K, m0, n0, N, acc0, acc1);

  const int lane = threadIdx.x & 31;
  const int nloc = lane & 15, hi = lane >> 4;
#pragma unroll
  for (int p = 0; p < 2; ++p) {
    v8f* acc = p ? acc1 : acc0;
#pragma unroll
    for (int v = 0; v < 8; ++v) {
      const int m = m0 + p * 16 + v + 8 * hi;  // C layout: VGPR v -> row v / v+8
      const float xm = xs2[m];
      float best = 3.0e38f;
#pragma unroll
      for (int t = 0; t < TNF; ++t) {
        int n = n0 + t * 16 + nloc;
        if (n < N) {
          float d2 = fmaxf(xm + cn2[n] - 2.0f * acc[t][v], 0.0f);
          best = fminf(best, d2);
        }
      }
#pragma unroll
      for (int o = 1; o < 16; o <<= 1) best = fminf(best, __shfl_xor(best, o, 32));
      if (nloc == 0) atomicMin(&mind2[m], __float_as_uint(best));
    }
  }
}

// MODE 0: C = A*W^T + bias ; MODE 1: tanh(A*W^T + bias) ; MODE 2: (A*W^T)*rs[m]*cs[n]
template <int MODE>
__global__ void gemm_xwT_epi_kernel(const float* __restrict__ A,
                                    const float* __restrict__ W,
                                    const float* __restrict__ zrow,
                                    int K, int N, int ldc,
                                    const float* __restrict__ bias,
                                    const float* __restrict__ rs,
                                    const float* __restrict__ cs,
                                    float* __restrict__ C) {
  int m0, n0;
  wave_coords(m0, n0);
  v8f acc0[TNF] = {}, acc1[TNF] = {};
  wave_gemm_rowrow2(A, K, W, K, zrow, K, m0, n0, N, acc0, acc1);

  const int lane = threadIdx.x & 31;
  const int nloc = lane & 15, hi = lane >> 4;
#pragma unroll
  for (int t = 0; t < TNF; ++t) {
    const int n = n0 + t * 16 + nloc;
    if (n < N) {
      const float coln = (MODE == 2) ? cs[n] : (bias ? bias[n] : 0.f);
#pragma unroll
      for (int p = 0; p < 2; ++p) {
        const v8f* acc = p ? acc1 : acc0;
#pragma unroll
        for (int v = 0; v < 8; ++v) {
          const int m = m0 + p * 16 + v + 8 * hi;
          float val = acc[t][v];
          if (MODE == 0) val += coln;
          if (MODE == 1) val = tanhf(val + coln);
          if (MODE == 2) val = val * rs[m] * coln;
          C[(long)m * ldc + n] = val;
        }
      }
    }
  }
}

// centers_cur = prob @ centers  (prob [B,K] row-major, centers [K,N] row-major)
__global__ void gemm_probB_kernel(const float* __restrict__ prob,
                                  const float* __restrict__ centers,
                                  int K, int N, float* __restrict__ C) {
  int m0, n0;
  wave_coords(m0, n0);
  v8f acc0[TNF] = {}, acc1[TNF] = {};
  wave_gemm_rowB2(prob, K, centers, N, K, m0, n0, acc0, acc1);

  const int lane = threadIdx.x & 31;
  const int nloc = lane & 15, hi = lane >> 4;
#pragma unroll
  for (int t = 0; t < TNF; ++t) {
    const int n = n0 + t * 16 + nloc;
#pragma unroll
    for (int p = 0; p < 2; ++p) {
      const v8f* acc = p ? acc1 : acc0;
#pragma unroll
      for (int v = 0; v < 8; ++v) {
        const int m = m0 + p * 16 + v + 8 * hi;
        C[(long)m * N + n] = acc[t][v];
      }
    }
  }
}

__global__ void softmax_kernel(const float* __restrict__ S, int C,
                               float* __restrict__ P) {
  const long si = (long)blockIdx.x * C;
  float mx = -3.0e38f;
  for (int i = threadIdx.x; i < C; i += blockDim.x) mx = fmaxf(mx, S[si + i]);
  mx = block_allreduce_max(mx);
  float sum = 0.f;
  for (int i = threadIdx.x; i < C; i += blockDim.x) {
    float e = expf(S[si + i] - mx);
    P[si + i] = e;
    sum += e;
  }
  sum = block_allreduce_sum(sum);
  const float inv = 1.0f / sum;
  for (int i = threadIdx.x; i < C; i += blockDim.x) P[si + i] *= inv;
}

// fast = gate*ccur ; x_out = conf*(x+fast) ; srow = 16/(1+||x_out||) ; emit slow/fast.
__global__ void fuse_kernel(const float* __restrict__ x,
                            const float* __restrict__ gate,
                            const float* __restrict__ ccur,
                            const unsigned* __restrict__ mind2, int D,
                            float* __restrict__ out_slow,
                            float* __restrict__ out_fast,
                            float* __restrict__ xout,
                            float* __restrict__ srow) {
  const int b = blockIdx.x;
  const float conf = 10.0f * rsqrtf(__uint_as_float(mind2[b]));  // 10 / min_dist
  const long off = (long)b * D;
  float ss = 0.f;
  for (int i = threadIdx.x; i < D; i += blockDim.x) {
    float xi = x[off + i];
    float ff = gate[off + i] * ccur[off + i];
    float xo = conf * (xi + ff);
    out_slow[off + i] = xi;
    out_fast[off + i] = ff;
    xout[off + i]     = xo;
    ss += xo * xo;
  }
  ss = block_allreduce_sum(ss);
  if (threadIdx.x == 0) srow[b] = 16.0f / (1.0f + sqrtf(ss));  // SCALE_COS/(1+norm)
}

// ---------------- launcher ----------------

extern "C" void kernel_launch(void* const* d_in, const int* in_sizes, int n_in,
                              void* d_out, int out_size, void* d_ws, size_t ws_size,
                              hipStream_t stream) {
  const float* x       = (const float*)d_in[0];
  // d_in[1] labels, d_in[3] class_count: unused by the forward outputs
  const float* centers = (const float*)d_in[2];
  const float* Wc      = (const float*)d_in[4];
  const float* bc      = (const float*)d_in[5];
  const float* W1      = (const float*)d_in[6];
  const float* b1      = (const float*)d_in[7];
  const float* Wcos    = (const float*)d_in[8];

  const int B = 128, D = 2048, C = 1000;  // C % 4 == 0 -> no K padding needed anywhere

  float* ws   = (float*)d_ws;
  float* S2   = ws; ws += (long)B * C;    // stage-1 logits
  float* prob = ws; ws += (long)B * C;    // softmax probabilities
  float* gate = ws; ws += (long)B * D;    // tanh gate
  float* ccur = ws; ws += (long)B * D;    // prob @ centers
  float* xout = ws; ws += (long)B * D;    // confidence*(x+fast)
  float* cn2  = ws; ws += C;              // ||center||^2
  float* winv = ws; ws += C;              // 1/||Wcos row||
  float* xs2  = ws; ws += B;              // ||x||^2
  float* srow = ws; ws += B;              // 16/(1+||x_out||)
  float* zrow = ws; ws += D;              // zeroed dummy row for OOB columns
  unsigned* mind2 = (unsigned*)ws;        // min dist^2 (uint bits)

  float* logits = (float*)d_out;
  float* slow   = logits + (long)B * C;
  float* fast   = slow + (long)B * D;

  const dim3 blk(256);                 // 8 wave32: 4 m-pairs x 2 n-groups -> 128x128/block
  const int gN_C = (C + 127) / 128;    // 8
  const int gN_D = D / 128;            // 16

  init_ws_kernel<<<(D + 255) / 256, blk, 0, stream>>>(zrow, D, mind2, B);
  rownorm_kernel<<<C, blk, 0, stream>>>(centers, D, 0, cn2);
  rownorm_kernel<<<C, blk, 0, stream>>>(Wcos, D, 1, winv);
  rownorm_kernel<<<B, blk, 0, stream>>>(x, D, 0, xs2);

  gemm_dist_min_kernel<<<gN_C, blk, 0, stream>>>(x, centers, zrow, D, C, xs2, cn2, mind2);
  gemm_xwT_epi_kernel<0><<<gN_C, blk, 0, stream>>>(x, W1, zrow, D, C, C, b1, nullptr, nullptr, S2);
  softmax_kernel<<<B, blk, 0, stream>>>(S2, C, prob);
  gemm_probB_kernel<<<gN_D, blk, 0, stream>>>(prob, centers, C, D, ccur);
  gemm_xwT_epi_kernel<1><<<gN_D, blk, 0, stream>>>(x, Wc, zrow, D, D, D, bc, nullptr, nullptr, gate);
  fuse_kernel<<<B, blk, 0, stream>>>(x, gate, ccur, mind2, D, slow, fast, xout, srow);
  gemm_xwT_epi_kernel<2><<<gN_C, blk, 0, stream>>>(xout, Wcos, zrow, D, C, C, nullptr, srow, winv, logits);
}